// GGNN_87917980549369
// MI455X (gfx1250) — compile-verified
//
#include <hip/hip_runtime.h>
#include <hip/hip_bf16.h>
#include <math.h>

#define N_NODES   100000
#define N_EDGES   1600000
#define HID       16
#define MSG       16
#define N_ETYPE   16
#define N_CLASSES 64

typedef float v2f __attribute__((ext_vector_type(2)));
typedef float v8f __attribute__((ext_vector_type(8)));

// ---------------------------------------------------------------------------
// Kernel 1: zero the message accumulator (must re-zero every call: atomics)
// ---------------------------------------------------------------------------
__global__ void ggnn_zero_kernel(float* __restrict__ p, int n) {
    int i = blockIdx.x * blockDim.x + threadIdx.x;
    if (i < n) p[i] = 0.0f;
}

// ---------------------------------------------------------------------------
// Kernel 2: per-edge message  msg = A[e_type] @ h_src, scatter-add into m[dst]
// edge_emb (16 KB) cached in LDS with stride-257 padding (257 mod 64 == 1 so
// the 16 edge types land on distinct banks for a given (k,j)).
// ---------------------------------------------------------------------------
__global__ __launch_bounds__(256) void ggnn_edge_msg_kernel(
    const float* __restrict__ feat, const int* __restrict__ src,
    const int* __restrict__ dst, const int* __restrict__ etype,
    const float* __restrict__ edge_emb, float* __restrict__ m) {
    __shared__ float As[N_ETYPE * 257];
    for (int i = threadIdx.x; i < N_ETYPE * 256; i += 256) {
        int t = i >> 8, o = i & 255;
        As[t * 257 + o] = edge_emb[i];
    }
    __syncthreads();

    int e = blockIdx.x * 256 + threadIdx.x;
    if (e >= N_EDGES) return;
    int s = src[e], d = dst[e], t = etype[e];

    const float4* hp = (const float4*)(feat + (long)s * HID);
    float4 h0 = hp[0], h1 = hp[1], h2 = hp[2], h3 = hp[3];

    const float* Ae = &As[t * 257];
    float* md = m + (long)d * MSG;
#pragma unroll
    for (int k = 0; k < MSG; ++k) {
        const float* r = Ae + k * HID;
        float acc = r[0]  * h0.x + r[1]  * h0.y + r[2]  * h0.z + r[3]  * h0.w
                  + r[4]  * h1.x + r[5]  * h1.y + r[6]  * h1.z + r[7]  * h1.w
                  + r[8]  * h2.x + r[9]  * h2.y + r[10] * h2.z + r[11] * h2.w
                  + r[12] * h3.x + r[13] * h3.y + r[14] * h3.z + r[15] * h3.w;
        unsafeAtomicAdd(md + k, acc);   // native global_atomic_add_f32
    }
}

// ---------------------------------------------------------------------------
// Kernel 3: GRU + output projection using V_WMMA_F32_16X16X4_F32.
// One wave32 per 16-node tile.
//
// ISA 7.12.2 layouts (f32):
//   A 16x4 : lanes 0-15 -> M=lane, K={kc,kc+1}; lanes 16-31 -> M=lane-16, K={kc+2,kc+3}
//   B 4x16 : VGPR r: lanes 0-15 -> K=kc+r, N=lane; lanes 16-31 -> K=kc+r+2, N=lane-16
//   C 16x16: VGPR r: lanes 0-15 -> M=r, N=lane;  lanes 16-31 -> M=r+8, N=lane-16
// ---------------------------------------------------------------------------
__device__ __forceinline__ v2f ggnn_a_frag(const float* __restrict__ X, int ld,
                                           int kc, int lane) {
    int M  = lane & 15;
    int kk = kc + ((lane >> 4) << 1);
    v2f a;
    a.x = X[M * ld + kk];
    a.y = X[M * ld + kk + 1];
    return a;
}

// B[k][n] = W[n][k]  (weights stored row-major [outCols][K], we need W^T)
__device__ __forceinline__ v2f ggnn_bT_frag(const float* __restrict__ W, int ld,
                                            int colBase, int kc, int lane) {
    int n  = colBase + (lane & 15);
    int kk = kc + ((lane >> 4) << 1);
    v2f b;
    b.x = W[n * ld + kk];
    b.y = W[n * ld + kk + 1];
    return b;
}

__device__ __forceinline__ float ggnn_sigmoid(float x) {
    return 1.0f / (1.0f + __expf(-x));
}

__global__ __launch_bounds__(256) void ggnn_gru_out_kernel(
    const float* __restrict__ feat, const float* __restrict__ m,
    const float* __restrict__ W_ih, const float* __restrict__ W_hh,
    const float* __restrict__ b_ih, const float* __restrict__ b_hh,
    const float* __restrict__ W_out, const float* __restrict__ b_out,
    float* __restrict__ out) {
    __shared__ float hstage[8][16 * 17];   // stride-17: conflict-free transpose

    const int lane = threadIdx.x & 31;
    const int wave = threadIdx.x >> 5;
    const int tile = blockIdx.x * 8 + wave;
    const int NT   = N_NODES / 16;         // 6250, exact
    if (tile >= NT) return;                // wave-uniform: EXEC stays all-ones

    const int base = tile * 16;
    const int col  = lane & 15;
    const int hi   = lane >> 4;

    // A-fragments for m-tile and feature-tile (K=16 -> 4 chunks of K=4)
    v2f am[4], af[4];
#pragma unroll
    for (int c = 0; c < 4; ++c) {
        am[c] = ggnn_a_frag(m    + (long)base * MSG, MSG, 4 * c, lane);
        af[c] = ggnn_a_frag(feat + (long)base * HID, HID, 4 * c, lane);
    }

    // gx = m @ W_ih^T, gh = feat @ W_hh^T  (3 N-tiles of 16: r,z,n gates)
    v8f gx[3], gh[3];
#pragma unroll
    for (int t = 0; t < 3; ++t) {
        v8f cx = {0.f, 0.f, 0.f, 0.f, 0.f, 0.f, 0.f, 0.f};
        v8f ch = {0.f, 0.f, 0.f, 0.f, 0.f, 0.f, 0.f, 0.f};
#pragma unroll
        for (int c = 0; c < 4; ++c) {
            v2f bx = ggnn_bT_frag(W_ih, MSG, 16 * t, 4 * c, lane);
            v2f bh = ggnn_bT_frag(W_hh, HID, 16 * t, 4 * c, lane);
            cx = __builtin_amdgcn_wmma_f32_16x16x4_f32(
                false, am[c], false, bx, (short)0, cx, false, false);
            ch = __builtin_amdgcn_wmma_f32_16x16x4_f32(
                false, af[c], false, bh, (short)0, ch, false, false);
        }
        gx[t] = cx;
        gh[t] = ch;
    }

    // GRU gate math per C-layout element (M = r + 8*hi, hidden index = col)
    float bir = b_ih[col],      bhr = b_hh[col];
    float biz = b_ih[16 + col], bhz = b_hh[16 + col];
    float bin = b_ih[32 + col], bhn = b_hh[32 + col];
    float* hs = hstage[wave];
#pragma unroll
    for (int r = 0; r < 8; ++r) {
        int   M  = r + 8 * hi;
        float rg = ggnn_sigmoid(gx[0][r] + bir + gh[0][r] + bhr);
        float zg = ggnn_sigmoid(gx[1][r] + biz + gh[1][r] + bhz);
        float ng = tanhf(gx[2][r] + bin + rg * (gh[2][r] + bhn));
        float hp = feat[(long)(base + M) * HID + col];
        hs[M * 17 + col] = (1.0f - zg) * ng + zg * hp;   // h_new
    }

    // cross-lane LDS dependence within the wave: drain DS counter
    asm volatile("s_wait_dscnt 0" ::: "memory");

    // reload h_new from LDS in A-matrix layout
    v2f ah[4];
#pragma unroll
    for (int c = 0; c < 4; ++c) {
        int kk = 4 * c + 2 * hi;
        v2f a;
        a.x = hs[col * 17 + kk];
        a.y = hs[col * 17 + kk + 1];
        ah[c] = a;
    }

    // out = h_new @ W_out^T + b_out   (4 N-tiles of 16 -> 64 classes)
#pragma unroll
    for (int t = 0; t < 4; ++t) {
        float bo = b_out[16 * t + col];
        v8f   cc = {bo, bo, bo, bo, bo, bo, bo, bo};
#pragma unroll
        for (int c = 0; c < 4; ++c) {
            v2f b = ggnn_bT_frag(W_out, HID, 16 * t, 4 * c, lane);
            cc = __builtin_amdgcn_wmma_f32_16x16x4_f32(
                false, ah[c], false, b, (short)0, cc, false, false);
        }
#pragma unroll
        for (int r = 0; r < 8; ++r) {
            int M = r + 8 * hi;
            out[(long)(base + M) * N_CLASSES + 16 * t + col] = cc[r];
        }
    }
}

// ---------------------------------------------------------------------------
extern "C" void kernel_launch(void* const* d_in, const int* in_sizes, int n_in,
                              void* d_out, int out_size, void* d_ws, size_t ws_size,
                              hipStream_t stream) {
    const float* feat     = (const float*)d_in[0];
    const int*   src      = (const int*)  d_in[1];
    const int*   dst      = (const int*)  d_in[2];
    const int*   etype    = (const int*)  d_in[3];
    const float* edge_emb = (const float*)d_in[4];
    const float* W_ih     = (const float*)d_in[5];
    const float* W_hh     = (const float*)d_in[6];
    const float* b_ih     = (const float*)d_in[7];
    const float* b_hh     = (const float*)d_in[8];
    const float* W_out    = (const float*)d_in[9];
    const float* b_out    = (const float*)d_in[10];
    float*       out      = (float*)d_out;
    float*       m        = (float*)d_ws;          // N_NODES*16 floats = 6.4 MB

    const int mElems = N_NODES * MSG;
    ggnn_zero_kernel<<<(mElems + 255) / 256, 256, 0, stream>>>(m, mElems);

    ggnn_edge_msg_kernel<<<(N_EDGES + 255) / 256, 256, 0, stream>>>(
        feat, src, dst, etype, edge_emb, m);

    const int nTiles = N_NODES / 16;               // 6250
    ggnn_gru_out_kernel<<<(nTiles + 7) / 8, 256, 0, stream>>>(
        feat, m, W_ih, W_hh, b_ih, b_hh, W_out, b_out, out);
}